// DistanceNetwork_12713103196210
// MI455X (gfx1250) — compile-verified
//
#include <hip/hip_runtime.h>

// out[j,b] = dot(input[j,:], support[j,b,:]) * rsqrt(max(||support[j,b,:]||^2, 1e-10))
// S = B = 1024, D = 256, fp32. Memory-bound: 1 GiB of support read once -> ~46us @ 23.3 TB/s.
// Dot products computed on the fp32 WMMA pipe (V_WMMA_F32_16X16X4_F32) with a broadcast-A
// trick; squared norms accumulated on VALU from the same loaded data (zero extra bytes).

typedef __attribute__((ext_vector_type(2))) float v2f;
typedef __attribute__((ext_vector_type(8))) float v8f;

#define S_DIM 1024
#define B_DIM 1024
#define D_DIM 256
#define WAVES_PER_BLOCK 8

__global__ __launch_bounds__(256) void distnet_wmma_f32(
    const float* __restrict__ support,   // [S, B, D]
    const float* __restrict__ input,     // [B(=S), D]  (row j used with support[j,:,:])
    float* __restrict__ out)             // [S, B]
{
    const int lane = threadIdx.x & 31;
    const int wave = threadIdx.x >> 5;
    const int tile = blockIdx.x * WAVES_PER_BLOCK + wave;   // 65536 tiles total
    const int j    = tile >> 6;                             // 0..1023
    const int b0   = (tile & 63) << 4;                      // 0,16,...,1008

    const int half = lane >> 4;    // 0: K-pair {0,1}, 1: K-pair {2,3} within each K=4 chunk
    const int col  = lane & 15;    // N = b offset within tile

    // B-matrix source: column n holds support row (b0+n). Per lane: 8B at k0 + 2*half.
    const float* supLane = support + ((size_t)j * B_DIM + (size_t)(b0 + col)) * D_DIM + 2 * half;
    // A-matrix source: all 16 rows identical = input[j] chunk (broadcast load, cache-hot).
    const float* inpLane = input + (size_t)j * D_DIM + 2 * half;

    v8f   acc = {};      // 16x16 f32 accumulator (8 VGPRs, wave32 layout)
    float sq  = 0.0f;    // per-lane partial squared norm for b = b0+col (this K-half)

#pragma unroll 4
    for (int k0 = 0; k0 < D_DIM; k0 += 4) {
        v2f b = *(const v2f*)(supLane + k0);   // K = k0+2h, k0+2h+1  -> B VGPRs v0,v1
        v2f a = *(const v2f*)(inpLane + k0);   // same K mapping      -> A VGPRs v0,v1
        sq += b.x * b.x + b.y * b.y;
        // D = A(16x4) x B(4x16) + C ; args: (neg_a, A, neg_b, B, c_mod, C, reuse_a, reuse_b)
        acc = __builtin_amdgcn_wmma_f32_16x16x4_f32(
            false, a, false, b, (short)0, acc, false, false);
    }

    // Combine the two K-half norm partials (lanes L and L^16 share the same b).
    sq += __shfl_xor(sq, 16, 32);
    const float inv = rsqrtf(fmaxf(sq, 1e-10f));

    // Every lane's acc[0] holds the full dot for b = b0+col (all D rows identical).
    // Lanes 0-15 do one coalesced 16-lane store; lanes 16-31 hold duplicates.
    if (lane < 16) {
        out[(size_t)j * B_DIM + (size_t)(b0 + col)] = acc[0] * inv;
    }
}

extern "C" void kernel_launch(void* const* d_in, const int* in_sizes, int n_in,
                              void* d_out, int out_size, void* d_ws, size_t ws_size,
                              hipStream_t stream) {
    const float* support = (const float*)d_in[0];   // S*B*D fp32
    const float* input   = (const float*)d_in[1];   // B*D fp32
    float* out = (float*)d_out;                     // S*B fp32

    const int tiles  = S_DIM * (B_DIM / 16);        // 65536
    const int blocks = tiles / WAVES_PER_BLOCK;     // 8192
    hipLaunchKernelGGL(distnet_wmma_f32, dim3(blocks), dim3(256), 0, stream,
                       support, input, out);
}